// RecSysGNN_18202071400770
// MI455X (gfx1250) — compile-verified
//
#include <hip/hip_runtime.h>
#include <hip/hip_bf16.h>
#include <math.h>

#define EMB_DIM 128

typedef float f4 __attribute__((ext_vector_type(4)));
typedef int   v4i __attribute__((vector_size(16)));  // matches async builtin param type

// ---------------- CDNA5 feature guards (compile-safe) ----------------
#if __has_builtin(__builtin_amdgcn_global_load_async_to_lds_b128)
#define USE_ASYNC_LDS 1
typedef __attribute__((address_space(1))) v4i as1_v4i;  // global (AS1) typed ptr
typedef __attribute__((address_space(3))) v4i as3_v4i;  // LDS (AS3) typed ptr
typedef __attribute__((address_space(3))) f4  as3_f4;
#else
#define USE_ASYNC_LDS 0
#endif

#if __has_builtin(__builtin_amdgcn_s_wait_asynccnt)
#define WAIT_ASYNC(n) __builtin_amdgcn_s_wait_asynccnt(n)
#else
#define WAIT_ASYNC(n) asm volatile("s_wait_asynccnt %0" ::"i"(n) : "memory")
#endif

// float atomic max via integer punning (works for any sign mix)
__device__ __forceinline__ void atomic_max_f32(float* addr, float v) {
  if (v >= 0.0f) {
    atomicMax((int*)addr, __float_as_int(v));
  } else {
    atomicMin((unsigned int*)addr, __float_as_uint(v));
  }
}

// ---------------- Phase 1: segment softmax stats ----------------
__global__ void k_init_stats(float* __restrict__ m_in, float* __restrict__ m_out,
                             float* __restrict__ s_in, float* __restrict__ s_out, int n) {
  int i = blockIdx.x * blockDim.x + threadIdx.x;
  if (i < n) {
    m_in[i]  = -INFINITY;
    m_out[i] = -INFINITY;
    s_in[i]  = 0.0f;
    s_out[i] = 0.0f;
  }
}

__global__ void k_edge_max(const float* __restrict__ attr, const int* __restrict__ from_,
                           const int* __restrict__ to_, float* __restrict__ m_in,
                           float* __restrict__ m_out, int E) {
  int e = blockIdx.x * blockDim.x + threadIdx.x;
  if (e < E) {
    float a = attr[e];
    atomic_max_f32(&m_in[to_[e]], a);
    atomic_max_f32(&m_out[from_[e]], a);
  }
}

__global__ void k_edge_expsum(const float* __restrict__ attr, const int* __restrict__ from_,
                              const int* __restrict__ to_, const float* __restrict__ m_in,
                              const float* __restrict__ m_out, float* __restrict__ s_in,
                              float* __restrict__ s_out, int E) {
  int e = blockIdx.x * blockDim.x + threadIdx.x;
  if (e < E) {
    float a = attr[e];
    int   t = to_[e], f = from_[e];
    atomicAdd(&s_in[t], __expf(a - m_in[t]));
    atomicAdd(&s_out[f], __expf(a - m_out[f]));
  }
}

__global__ void k_edge_norm(const float* __restrict__ attr, const int* __restrict__ from_,
                            const int* __restrict__ to_, const float* __restrict__ m_in,
                            const float* __restrict__ m_out, const float* __restrict__ s_in,
                            const float* __restrict__ s_out, float* __restrict__ norm, int E) {
  int e = blockIdx.x * blockDim.x + threadIdx.x;
  if (e < E) {
    float a   = attr[e];
    int   t   = to_[e], f = from_[e];
    float pin  = __expf(a - m_in[t]) / s_in[t];
    float pout = __expf(a - m_out[f]) / s_out[f];
    norm[e]    = sqrtf(pin * pout);
  }
}

// ---------------- Phase 2: embedding init (vectorized b128) ----------------
// out0 = emb ; acc = emb/4 ; A = emb (layer-0 source) ; B = 0 (layer-1 dest)
__global__ void k_init_emb(const f4* __restrict__ emb, f4* __restrict__ out0,
                           f4* __restrict__ acc, f4* __restrict__ A,
                           f4* __restrict__ B, int n4) {
  int i = blockIdx.x * blockDim.x + threadIdx.x;
  if (i < n4) {
    f4 v = emb[i];
    out0[i] = v;
    acc[i]  = v * 0.25f;
    A[i]    = v;
    f4 z = {0.0f, 0.0f, 0.0f, 0.0f};
    B[i] = z;
  }
}

// ---------------- Phase 3: one layer of propagation ----------------
// One wave (32 lanes) per edge: 32 lanes x float4 == one 128-float row.
// Gather is double-buffered through LDS via GLOBAL_LOAD_ASYNC_TO_LDS_B128
// (ASYNCcnt-pipelined), with a depth-2 global_prefetch_b8 into L2.
__global__ void k_propagate(const float* __restrict__ src, float* __restrict__ dst,
                            const int* __restrict__ from_, const int* __restrict__ to_,
                            const float* __restrict__ norm, int E) {
  const int lane = threadIdx.x & 31;
  const int wv   = threadIdx.x >> 5;
  const int gw   = blockIdx.x * (blockDim.x >> 5) + wv;
  const int nw   = gridDim.x * (blockDim.x >> 5);

#if USE_ASYNC_LDS
  __shared__ f4 stage[8][2][32];  // blockDim.x == 256 -> 8 waves, 2-deep pipeline
  int parity = 0;
  if (gw < E) {
    const float* p0 = src + (size_t)from_[gw] * EMB_DIM + lane * 4;
    __builtin_amdgcn_global_load_async_to_lds_b128(
        (as1_v4i*)p0, (as3_v4i*)&stage[wv][0][lane], 0, 0);
  }
  for (int e = gw; e < E; e += nw) {
    int en = e + nw;
    // hoisted loads: overlap with the in-flight async fill
    float s = norm[e];
    int   t = to_[e];
    if (en < E) {
      const float* pn = src + (size_t)from_[en] * EMB_DIM + lane * 4;
      __builtin_amdgcn_global_load_async_to_lds_b128(
          (as1_v4i*)pn, (as3_v4i*)&stage[wv][parity ^ 1][lane], 0, 0);
      int e2 = en + nw;
      if (e2 < E) __builtin_prefetch(src + (size_t)from_[e2] * EMB_DIM + lane * 4, 0, 1);
      WAIT_ASYNC(1);
    } else {
      WAIT_ASYNC(0);
    }
    // launder the LDS address so the compiler can't reason about the (HW-written) staging buffer
    unsigned loff = (unsigned)(size_t)(as3_f4*)&stage[wv][parity][lane];
    asm volatile("" : "+v"(loff)::"memory");
    f4 v = *(as3_f4*)(size_t)loff;

    float* d = dst + (size_t)t * EMB_DIM + lane * 4;
    atomicAdd(d + 0, s * v.x);
    atomicAdd(d + 1, s * v.y);
    atomicAdd(d + 2, s * v.z);
    atomicAdd(d + 3, s * v.w);
    parity ^= 1;
  }
#else
  for (int e = gw; e < E; e += nw) {
    int en = e + nw;
    if (en < E) __builtin_prefetch(src + (size_t)from_[en] * EMB_DIM + lane * 4, 0, 1);
    const f4* row = (const f4*)(src + (size_t)from_[e] * EMB_DIM);
    f4 v = row[lane];
    float s  = norm[e];
    float* d = dst + (size_t)to_[e] * EMB_DIM + lane * 4;
    atomicAdd(d + 0, s * v.x);
    atomicAdd(d + 1, s * v.y);
    atomicAdd(d + 2, s * v.z);
    atomicAdd(d + 3, s * v.w);
  }
#endif
}

// ---------------- Phase 4: acc += res/4 ; zero the other ping-pong buffer ----------------
__global__ void k_acc_zero(f4* __restrict__ acc, const f4* __restrict__ res,
                           f4* __restrict__ zero_me, int n4) {
  int i = blockIdx.x * blockDim.x + threadIdx.x;
  if (i < n4) {
    acc[i] += res[i] * 0.25f;
    f4 z = {0.0f, 0.0f, 0.0f, 0.0f};
    zero_me[i] = z;
  }
}

__global__ void k_acc_final(f4* __restrict__ acc, const f4* __restrict__ res, int n4) {
  int i = blockIdx.x * blockDim.x + threadIdx.x;
  if (i < n4) {
    acc[i] += res[i] * 0.25f;
  }
}

extern "C" void kernel_launch(void* const* d_in, const int* in_sizes, int n_in,
                              void* d_out, int out_size, void* d_ws, size_t ws_size,
                              hipStream_t stream) {
  const float* emb  = (const float*)d_in[0];
  const int*   ei   = (const int*)d_in[1];
  const float* attr = (const float*)d_in[2];

  const int nodes = in_sizes[0] / EMB_DIM;  // 100000
  const int E     = in_sizes[2];            // 600000
  const int* from_ = ei;
  const int* to_   = ei + E;

  float* out0 = (float*)d_out;                   // emb0 copy: nodes*128
  float* acc  = out0 + (size_t)nodes * EMB_DIM;  // layer-average accumulator

  // workspace carving (element counts here are multiples of 4 -> 16B aligned)
  float* p     = (float*)d_ws;
  float* m_in  = p; p += nodes;
  float* m_out = p; p += nodes;
  float* s_in  = p; p += nodes;
  float* s_out = p; p += nodes;
  float* norm  = p; p += E;
  size_t up = ((size_t)p + 15) & ~(size_t)15;
  p = (float*)up;
  float* A = p; p += (size_t)nodes * EMB_DIM;  // ping
  float* B = p;                                // pong

  const dim3 tb(256);
  const int nb_nodes = (nodes + 255) / 256;
  const int nb_edges = (E + 255) / 256;
  const int n4       = nodes * (EMB_DIM / 4);
  const int nb_n4    = (n4 + 255) / 256;

  // edge-softmax normalization
  k_init_stats<<<nb_nodes, tb, 0, stream>>>(m_in, m_out, s_in, s_out, nodes);
  k_edge_max<<<nb_edges, tb, 0, stream>>>(attr, from_, to_, m_in, m_out, E);
  k_edge_expsum<<<nb_edges, tb, 0, stream>>>(attr, from_, to_, m_in, m_out, s_in, s_out, E);
  k_edge_norm<<<nb_edges, tb, 0, stream>>>(attr, from_, to_, m_in, m_out, s_in, s_out, norm, E);

  // emb0 copy + accumulator init + ping-pong init
  k_init_emb<<<nb_n4, tb, 0, stream>>>((const f4*)emb, (f4*)out0, (f4*)acc, (f4*)A, (f4*)B, n4);

  // layer 1: A -> B ; acc += B/4 ; zero A (next dest)
  k_propagate<<<4096, tb, 0, stream>>>(A, B, from_, to_, norm, E);
  k_acc_zero<<<nb_n4, tb, 0, stream>>>((f4*)acc, (const f4*)B, (f4*)A, n4);
  // layer 2: B -> A ; acc += A/4 ; zero B (next dest)
  k_propagate<<<4096, tb, 0, stream>>>(B, A, from_, to_, norm, E);
  k_acc_zero<<<nb_n4, tb, 0, stream>>>((f4*)acc, (const f4*)A, (f4*)B, n4);
  // layer 3: A -> B ; acc += B/4 (no more zeroing needed)
  k_propagate<<<4096, tb, 0, stream>>>(A, B, from_, to_, norm, E);
  k_acc_final<<<nb_n4, tb, 0, stream>>>((f4*)acc, (const f4*)B, n4);
}